// GAT_40870908789103
// MI455X (gfx1250) — compile-verified
//
#include <hip/hip_runtime.h>
#include <math.h>

typedef __attribute__((ext_vector_type(2))) float v2f;
typedef __attribute__((ext_vector_type(8))) float v8f;

#define GAT_FIN 128
#define GAT_H   8
#define GAT_O   8
#define GAT_C   64          // H*O
#define BSTRIDE 130         // padded LDS stride (conflict-free column reads)

// float atomic max via signed/unsigned integer monotonicity trick
__device__ __forceinline__ void atomicMaxF32(float* addr, float val) {
  if (val >= 0.0f) {
    atomicMax((int*)addr, __float_as_int(val));
  } else {
    atomicMin((unsigned int*)addr, __float_as_uint(val));
  }
}

// ---------------------------------------------------------------------------
// Kernel 1: Wh[n, h*8+o] = sum_f h[n,f] * W[h,f,o]  via V_WMMA_F32_16X16X4_F32
// One wave -> 16 rows x 64 cols. B (128x64 = 32KB) staged in LDS transposed.
// ---------------------------------------------------------------------------
__global__ __launch_bounds__(256) void gat_gemm_wmma(
    const float* __restrict__ h, const float* __restrict__ W,
    float* __restrict__ Wh, int nStrips)
{
  __shared__ float Bl[GAT_C * BSTRIDE];
  const int tid = threadIdx.x;
  // Bl[c][f] = W[c/8][f][c%8]   (B transposed: column-major-ish for ds_load_b64)
  for (int idx = tid; idx < GAT_FIN * GAT_C; idx += 256) {
    int f = idx >> 6;
    int c = idx & 63;
    Bl[c * BSTRIDE + f] = W[((c >> 3) * GAT_FIN + f) * GAT_O + (c & 7)];
  }
  __syncthreads();

  const int wave  = tid >> 5;
  const int lane  = tid & 31;
  const int col   = lane & 15;          // M for A-frag, N for B/C-frag
  const int kbase = (lane >> 4) << 1;   // 0 or 2: K-pair select
  const int strip = blockIdx.x * 8 + wave;
  if (strip >= nStrips) return;         // wave-uniform: EXEC all-ones inside

  const long row0 = (long)strip * 16;
  const float* hrow = h + (row0 + col) * GAT_FIN + kbase;
  const float* b0   = &Bl[col * BSTRIDE + kbase];

  v8f acc0 = {}, acc1 = {}, acc2 = {}, acc3 = {};
#pragma unroll
  for (int k0 = 0; k0 < GAT_FIN; k0 += 4) {
    v2f a  = *(const v2f*)(hrow + k0);
    v2f bA = *(const v2f*)(b0 + k0);
    v2f bB = *(const v2f*)(b0 + 16 * BSTRIDE + k0);
    v2f bC = *(const v2f*)(b0 + 32 * BSTRIDE + k0);
    v2f bD = *(const v2f*)(b0 + 48 * BSTRIDE + k0);
    acc0 = __builtin_amdgcn_wmma_f32_16x16x4_f32(false, a, false, bA, (short)0, acc0, false, false);
    acc1 = __builtin_amdgcn_wmma_f32_16x16x4_f32(false, a, false, bB, (short)0, acc1, false, false);
    acc2 = __builtin_amdgcn_wmma_f32_16x16x4_f32(false, a, false, bC, (short)0, acc2, false, false);
    acc3 = __builtin_amdgcn_wmma_f32_16x16x4_f32(false, a, false, bD, (short)0, acc3, false, false);
  }

  // C/D layout: VGPR i holds row i (lanes 0-15) / row i+8 (lanes 16-31), col = lane&15
  const int mofs = (lane >> 4) * 8;
  float* obase = Wh + row0 * GAT_C;
#pragma unroll
  for (int i = 0; i < 8; ++i) {
    float* r = obase + (long)(i + mofs) * GAT_C + col;
    r[0]  = acc0[i];
    r[16] = acc1[i];
    r[32] = acc2[i];
    r[48] = acc3[i];
  }
}

// ---------------------------------------------------------------------------
// Kernel 2: per-(node,head) attention half-scores
// ---------------------------------------------------------------------------
__global__ __launch_bounds__(256) void gat_scores(
    const float* __restrict__ Wh, const float* __restrict__ a1,
    const float* __restrict__ a2, float* __restrict__ s_src,
    float* __restrict__ s_dst, int NH)
{
  int i = blockIdx.x * 256 + threadIdx.x;   // n*8 + hd
  if (i >= NH) return;
  int hd  = i & 7;
  long n  = i >> 3;
  const float* w  = Wh + n * GAT_C + hd * GAT_O;
  const float* p1 = a1 + hd * GAT_O;
  const float* p2 = a2 + hd * GAT_O;
  float s1 = 0.f, s2 = 0.f;
#pragma unroll
  for (int o = 0; o < GAT_O; ++o) { float v = w[o]; s1 += v * p1[o]; s2 += v * p2[o]; }
  s_src[i] = s1;
  s_dst[i] = s2;
}

// ---------------------------------------------------------------------------
// Kernel 3: init m = -inf, denom = 0, out = 0
// ---------------------------------------------------------------------------
__global__ __launch_bounds__(256) void gat_init(
    float* __restrict__ m, float* __restrict__ denom,
    float* __restrict__ out, int NH, int NC)
{
  int i = blockIdx.x * 256 + threadIdx.x;
  if (i < NH) { m[i] = -INFINITY; denom[i] = 0.0f; }
  if (i < NC) out[i] = 0.0f;
}

// ---------------------------------------------------------------------------
// Kernel 4: edge logits + segment max   (thread per (edge, head))
// ---------------------------------------------------------------------------
__global__ __launch_bounds__(256) void gat_edge_logits(
    const int* __restrict__ src, const int* __restrict__ dst,
    const float* __restrict__ s_src, const float* __restrict__ s_dst,
    float* __restrict__ ew, float* __restrict__ m, long long E8)
{
  long long i = (long long)blockIdx.x * 256 + threadIdx.x;  // e*8 + hd
  if (i >= E8) return;
  int hd = (int)(i & 7);
  long long e = i >> 3;
  int s = src[e], d = dst[e];
  float v = s_src[(long)s * 8 + hd] + s_dst[(long)d * 8 + hd];
  v = v > 0.0f ? v : 0.2f * v;           // leaky_relu(0.2)
  ew[i] = v;
  atomicMaxF32(&m[(long)d * 8 + hd], v);
}

// ---------------------------------------------------------------------------
// Kernel 5: exp(e - m[dst]) + segment sum
// ---------------------------------------------------------------------------
__global__ __launch_bounds__(256) void gat_edge_exp(
    const int* __restrict__ dst, const float* __restrict__ m,
    float* __restrict__ ew, float* __restrict__ denom, long long E8)
{
  long long i = (long long)blockIdx.x * 256 + threadIdx.x;
  if (i >= E8) return;
  int hd = (int)(i & 7);
  long long e = i >> 3;
  int d = dst[e];
  float mm = m[(long)d * 8 + hd];
  if (!__builtin_isfinite(mm)) mm = 0.0f;
  float p = __expf(ew[i] - mm);
  ew[i] = p;
  atomicAdd(&denom[(long)d * 8 + hd], p);
}

// ---------------------------------------------------------------------------
// Kernel 6: normalized weighted scatter-sum of messages
// ---------------------------------------------------------------------------
__global__ __launch_bounds__(256) void gat_edge_scatter(
    const int* __restrict__ src, const int* __restrict__ dst,
    const float* __restrict__ ew, const float* __restrict__ denom,
    const float* __restrict__ Wh, float* __restrict__ out, long long E8)
{
  long long i = (long long)blockIdx.x * 256 + threadIdx.x;
  if (i >= E8) return;
  int hd = (int)(i & 7);
  long long e = i >> 3;
  int s = src[e], d = dst[e];
  float a = ew[i] / fmaxf(denom[(long)d * 8 + hd], 1e-16f);
  const float* w = Wh + (long)s * GAT_C + hd * GAT_O;
  float* op = out + (long)d * GAT_C + hd * GAT_O;
#pragma unroll
  for (int o = 0; o < GAT_O; ++o) atomicAdd(&op[o], a * w[o]);
}

// ---------------------------------------------------------------------------
extern "C" void kernel_launch(void* const* d_in, const int* in_sizes, int n_in,
                              void* d_out, int out_size, void* d_ws, size_t ws_size,
                              hipStream_t stream)
{
  const float* h  = (const float*)d_in[0];
  const float* W  = (const float*)d_in[1];
  const float* a1 = (const float*)d_in[2];
  const float* a2 = (const float*)d_in[3];
  const int* src  = (const int*)d_in[4];
  const int* dst  = (const int*)d_in[5];
  float* out = (float*)d_out;

  const int N = in_sizes[0] / GAT_FIN;   // 50000
  const int E = in_sizes[4];             // 800000

  // workspace carve-up (floats)
  float* ws    = (float*)d_ws;
  float* Wh    = ws;                                  // N*64
  float* s_src = Wh    + (size_t)N * GAT_C;           // N*8
  float* s_dst = s_src + (size_t)N * GAT_H;           // N*8
  float* m     = s_dst + (size_t)N * GAT_H;           // N*8
  float* denom = m     + (size_t)N * GAT_H;           // N*8
  float* ew    = denom + (size_t)N * GAT_H;           // E*8

  const int nStrips = (N + 15) / 16;                  // 3125 (N % 16 == 0)
  const int gemmBlk = (nStrips + 7) / 8;
  gat_gemm_wmma<<<gemmBlk, 256, 0, stream>>>(h, W, Wh, nStrips);

  const int NH = N * GAT_H;
  gat_scores<<<(NH + 255) / 256, 256, 0, stream>>>(Wh, a1, a2, s_src, s_dst, NH);

  const int NC = N * GAT_C;
  gat_init<<<(NC + 255) / 256, 256, 0, stream>>>(m, denom, out, NH, NC);

  const long long E8 = (long long)E * GAT_H;
  const int eb = (int)((E8 + 255) / 256);
  gat_edge_logits<<<eb, 256, 0, stream>>>(src, dst, s_src, s_dst, ew, m, E8);
  gat_edge_exp<<<eb, 256, 0, stream>>>(dst, m, ew, denom, E8);
  gat_edge_scatter<<<eb, 256, 0, stream>>>(src, dst, ew, denom, Wh, out, E8);
}